// WhisperSelfAttentionStepTVM_76227079569643
// MI455X (gfx1250) — compile-verified
//
#include <hip/hip_runtime.h>

typedef __attribute__((ext_vector_type(2))) float v2f;
typedef __attribute__((ext_vector_type(8))) float v8f;

#define E  1280
#define H  20
#define L  8192
#define D  64
#define HLD (H * L * D)      // 10485760
#define NSPLIT 16
#define CHUNK (L / NSPLIT)   // 512
#define PSTRIDE (2 + D)      // m, z, acc[64] per (head, split)

// ---------------------------------------------------------------------------
// GEMV via V_WMMA_F32_16X16X4_F32:  y[n] = (sum_k x[k]*W[n*K+k] + b[n]) * scale
// One wave -> 16 output rows; block = 128 threads (4 waves) -> 64 rows.
// A (16x4):  lanes 0-15 hold M=0..15 with K=0,1 in v[0],v[1];
//            lanes 16-31 hold K=2,3.  Only row M=0 (lanes 0 & 16) is live.
// B (4x16):  lanes 0-15 N=lane with K=0,1; lanes 16-31 N=lane-16 with K=2,3.
// C/D row M=0 lands in c[0] of lanes 0-15.
// ---------------------------------------------------------------------------
__global__ void gemv_wmma_kernel(const float* __restrict__ x,
                                 const float* __restrict__ W,
                                 const float* __restrict__ bias,
                                 float* __restrict__ y,
                                 float scale, int has_bias)
{
    __shared__ float sx[E];
    for (int i = threadIdx.x; i < E; i += blockDim.x) sx[i] = x[i];
    __syncthreads();

    const int lane  = threadIdx.x & 31;
    const int wave  = threadIdx.x >> 5;
    const int n0    = (blockIdx.x * 4 + wave) * 16;
    const int n     = lane & 15;
    const int khalf = (lane >> 4) << 1;          // 0 or 2
    const bool row0 = (n == 0);                  // lanes carrying A row M=0

    const float* Wrow = W + (size_t)(n0 + n) * E;

    v8f c = {};
    for (int k0 = 0; k0 < E; k0 += 4) {
        v2f a, b;
        a.x = row0 ? sx[k0 + khalf]     : 0.0f;
        a.y = row0 ? sx[k0 + khalf + 1] : 0.0f;
        b.x = Wrow[k0 + khalf];
        b.y = Wrow[k0 + khalf + 1];
        c = __builtin_amdgcn_wmma_f32_16x16x4_f32(false, a, false, b,
                                                  (short)0, c, false, false);
    }
    if (lane < 16) {
        float r = c[0];
        if (has_bias) r += bias[n0 + lane];
        y[n0 + lane] = r * scale;
    }
}

// ---------------------------------------------------------------------------
// Bulk cache copy (~168 MB) via the CDNA5 async global<->LDS path:
// GLOBAL_LOAD_ASYNC_TO_LDS_B128 / GLOBAL_STORE_ASYNC_FROM_LDS_B128,
// fenced with ASYNCcnt. Data never touches VGPRs.
// LDS offset = low 32 bits of the generic pointer (flat-LDS mapping).
// ---------------------------------------------------------------------------
__global__ void cache_copy_async_kernel(const float* __restrict__ k_cache,
                                        const float* __restrict__ v_cache,
                                        float* __restrict__ k_out,
                                        float* __restrict__ v_out)
{
    __shared__ float4 stage[512];                 // 8 KB: [0,256)=K slots, [256,512)=V slots
    const unsigned lds_k = (unsigned)(uintptr_t)&stage[threadIdx.x];
    const unsigned lds_v = (unsigned)(uintptr_t)&stage[256 + threadIdx.x];

    const size_t n4 = (size_t)HLD / 4;
    const size_t stride = (size_t)gridDim.x * blockDim.x;
    for (size_t i4 = (size_t)blockIdx.x * blockDim.x + threadIdx.x; i4 < n4; i4 += stride) {
        const unsigned long long ksrc = (unsigned long long)(uintptr_t)(k_cache + i4 * 4);
        const unsigned long long vsrc = (unsigned long long)(uintptr_t)(v_cache + i4 * 4);
        const unsigned long long kdst = (unsigned long long)(uintptr_t)(k_out   + i4 * 4);
        const unsigned long long vdst = (unsigned long long)(uintptr_t)(v_out   + i4 * 4);

        asm volatile("global_load_async_to_lds_b128 %0, %1, off"
                     :: "v"(lds_k), "v"(ksrc) : "memory");
        asm volatile("global_load_async_to_lds_b128 %0, %1, off"
                     :: "v"(lds_v), "v"(vsrc) : "memory");
        asm volatile("s_wait_asynccnt 0x0" ::: "memory");
        asm volatile("global_store_async_from_lds_b128 %0, %1, off"
                     :: "v"(kdst), "v"(lds_k) : "memory");
        asm volatile("global_store_async_from_lds_b128 %0, %1, off"
                     :: "v"(vdst), "v"(lds_v) : "memory");
        asm volatile("s_wait_asynccnt 0x0" ::: "memory");
    }
}

// Overwrite row `pos` of both output caches with k_new / v_new (2*E floats).
__global__ void cache_fixup_kernel(const float* __restrict__ kv_new,  // k_new@0, v_new@E
                                   const int* __restrict__ pos_ptr,
                                   float* __restrict__ k_out,
                                   float* __restrict__ v_out)
{
    const int pos = pos_ptr[0];
    const int idx = blockIdx.x * blockDim.x + threadIdx.x;   // [0, 2E)
    if (idx < E) {
        const int h = idx >> 6, d = idx & (D - 1);
        k_out[((size_t)h * L + pos) * D + d] = kv_new[idx];
    } else if (idx < 2 * E) {
        const int r = idx - E;
        const int h = r >> 6, d = r & (D - 1);
        v_out[((size_t)h * L + pos) * D + d] = kv_new[E + r];
    }
}

// ---------------------------------------------------------------------------
// Flash-decoding partial attention: grid (H, NSPLIT), 256 threads.
// Per (head, split): local max m, local Z = sum exp(s-m), partial
// acc[d] = sum exp(s-m)*v[l,d] over l in the split (l <= pos only).
// ---------------------------------------------------------------------------
__global__ void attention_partial_kernel(const float* __restrict__ q,       // [H*D], pre-scaled
                                         const float* __restrict__ kv_new,  // k_new@0, v_new@E
                                         const float* __restrict__ k_cache,
                                         const float* __restrict__ v_cache,
                                         const int* __restrict__ pos_ptr,
                                         float* __restrict__ part)          // [H*NSPLIT*PSTRIDE]
{
    __shared__ float s_e[CHUNK];
    __shared__ float s_q[D];
    __shared__ float red[256];

    const int h = blockIdx.x, sp = blockIdx.y, tid = threadIdx.x;
    const int pos = pos_ptr[0];
    const int l0 = sp * CHUNK;
    const int lend = min(l0 + CHUNK - 1, pos);   // inclusive; may be < l0 (empty)
    float* Pp = part + (size_t)(h * NSPLIT + sp) * PSTRIDE;

    if (tid < D) s_q[tid] = q[h * D + tid];
    __syncthreads();

    const float* Kh = k_cache + (size_t)h * L * D;
    const float* kn = kv_new + h * D;
    float lmax = -1e30f;
    for (int l = l0 + tid; l <= lend; l += 256) {
        const float* kr = (l == pos) ? kn : (Kh + (size_t)l * D);
        float s = 0.f;
        #pragma unroll
        for (int j = 0; j < D; j += 4) {
            float4 k4 = *(const float4*)(kr + j);
            s += s_q[j] * k4.x + s_q[j + 1] * k4.y + s_q[j + 2] * k4.z + s_q[j + 3] * k4.w;
        }
        s_e[l - l0] = s;
        lmax = fmaxf(lmax, s);
    }
    red[tid] = lmax; __syncthreads();
    for (int off = 128; off > 0; off >>= 1) {
        if (tid < off) red[tid] = fmaxf(red[tid], red[tid + off]);
        __syncthreads();
    }
    const float m = red[0]; __syncthreads();

    float lsum = 0.f;
    for (int l = l0 + tid; l <= lend; l += 256) {
        float e = __expf(s_e[l - l0] - m);
        s_e[l - l0] = e;
        lsum += e;
    }
    red[tid] = lsum; __syncthreads();
    for (int off = 128; off > 0; off >>= 1) {
        if (tid < off) red[tid] += red[tid + off];
        __syncthreads();
    }
    const float z = red[0]; __syncthreads();

    const float* Vh = v_cache + (size_t)h * L * D;
    const float* vn = kv_new + E + h * D;
    const int d = tid & (D - 1);
    const int grp = tid >> 6;                    // 0..3
    float acc = 0.f;
    for (int l = l0 + grp; l <= lend; l += 4) {
        const float* vr = (l == pos) ? vn : (Vh + (size_t)l * D);
        acc += s_e[l - l0] * vr[d];
    }
    red[tid] = acc; __syncthreads();
    if (tid < D) {
        Pp[2 + tid] = red[tid] + red[tid + 64] + red[tid + 128] + red[tid + 192];
        if (tid == 0) { Pp[0] = m; Pp[1] = z; }
    }
}

// Combine split partials:  out[h,d] = sum_s acc_s[d]*exp(m_s-M) / sum_s z_s*exp(m_s-M)
__global__ void attn_combine_kernel(const float* __restrict__ part,
                                    float* __restrict__ a_out)
{
    const int h = blockIdx.x, d = threadIdx.x;   // 64 threads
    const float* P = part + (size_t)h * NSPLIT * PSTRIDE;
    float M = -1e30f;
    for (int s = 0; s < NSPLIT; ++s) M = fmaxf(M, P[s * PSTRIDE]);
    float Z = 0.f, acc = 0.f;
    for (int s = 0; s < NSPLIT; ++s) {
        const float w = __expf(P[s * PSTRIDE] - M);
        Z   += P[s * PSTRIDE + 1] * w;
        acc += P[s * PSTRIDE + 2 + d] * w;
    }
    a_out[h * D + d] = acc / Z;
}

// ---------------------------------------------------------------------------
extern "C" void kernel_launch(void* const* d_in, const int* in_sizes, int n_in,
                              void* d_out, int out_size, void* d_ws, size_t ws_size,
                              hipStream_t stream) {
    (void)in_sizes; (void)n_in; (void)out_size; (void)ws_size;
    const float* hs  = (const float*)d_in[0];
    const int*   pos = (const int*)d_in[1];   // int64 in memory; low word is the value
    const float* kc  = (const float*)d_in[2];
    const float* vc  = (const float*)d_in[3];
    const float* Wq  = (const float*)d_in[4];
    const float* bq  = (const float*)d_in[5];
    const float* Wk  = (const float*)d_in[6];
    const float* Wv  = (const float*)d_in[7];
    const float* bv  = (const float*)d_in[8];
    const float* Wo  = (const float*)d_in[9];
    const float* bo  = (const float*)d_in[10];

    float* out   = (float*)d_out;      // [E]
    float* k_out = out + E;            // [H*L*D]
    float* v_out = k_out + HLD;        // [H*L*D]

    float* ws    = (float*)d_ws;
    float* q_s   = ws;                 // E   (scaled q)
    float* kvnew = ws + E;             // 2E  (k_new, v_new)
    float* a_out = ws + 3 * E;         // E   (attention output, head-major)
    float* part  = ws + 4 * E;         // H*NSPLIT*PSTRIDE partials

    const dim3 blk(128);
    const dim3 grd(E / 64);            // 20 blocks, 4 waves * 16 rows each
    const float scale = 0.125f;        // D^-0.5

    // q = (h @ Wq.T + bq) * scale ; k = h @ Wk.T ; v = h @ Wv.T + bv
    gemv_wmma_kernel<<<grd, blk, 0, stream>>>(hs, Wq, bq, q_s,       scale, 1);
    gemv_wmma_kernel<<<grd, blk, 0, stream>>>(hs, Wk, bq, kvnew,     1.0f,  0);
    gemv_wmma_kernel<<<grd, blk, 0, stream>>>(hs, Wv, bv, kvnew + E, 1.0f,  1);

    // Bulk copy via async LDS path, then patch row `pos` with k_new/v_new.
    cache_copy_async_kernel<<<2048, 256, 0, stream>>>(kc, vc, k_out, v_out);
    cache_fixup_kernel<<<10, 256, 0, stream>>>(kvnew, pos, k_out, v_out);

    attention_partial_kernel<<<dim3(H, NSPLIT), 256, 0, stream>>>(q_s, kvnew, kc, vc, pos, part);
    attn_combine_kernel<<<H, D, 0, stream>>>(part, a_out);

    // out = attn_out @ Wo.T + bo
    gemv_wmma_kernel<<<grd, blk, 0, stream>>>(a_out, Wo, bo, out, 1.0f, 1);
}